// PointCloudAE_32014686224772
// MI455X (gfx1250) — compile-verified
//
#include <hip/hip_runtime.h>
#include <hip/hip_bf16.h>
#include <float.h>

typedef __attribute__((ext_vector_type(2))) float v2f;
typedef __attribute__((ext_vector_type(8))) float v8f;

#define EPS_BN 1e-5f

// ======================================================================
// GEMM + folded BN + ReLU via V_WMMA_F32_16X16X4_F32 (wave32).
// One wave computes a 16 x (16*TN) tile of Y[b] = relu(bn(W @ X[b])).
//   - P is guaranteed (by the launcher) to be a multiple of 16*TN -> no P guards.
//   - FULL_O: O % 16 == 0 -> no O guards at all (true for all layers but 32->6).
//   - C handled as guard-free main loop over C&~3 plus one guarded tail step.
// grid = (P/(16*TN), ceil(O/16), B), block = 32 (one wave)
// ======================================================================
template <int TN, bool FULL_O>
__global__ __launch_bounds__(32)
void k_conv_bn_relu(const float* __restrict__ X, float* __restrict__ Y,
                    const float* __restrict__ Wm, const float* __restrict__ bias,
                    const float* __restrict__ beta, const float* __restrict__ gamma,
                    int C, int O, int P)
{
    const int lane = threadIdx.x;       // 0..31
    const int l    = lane & 15;
    const int half = lane >> 4;         // K-pair selector for A/B operands
    const int o0   = blockIdx.y * 16;
    const int p0   = blockIdx.x * (16 * TN);
    const float* Xb = X + (size_t)blockIdx.z * (size_t)C * (size_t)P;
    float*       Yb = Y + (size_t)blockIdx.z * (size_t)O * (size_t)P;

    const int ar = o0 + l;              // A row (output channel) for this lane
    const int pc = p0 + l;              // first B col (point) for this lane

    // Branch-free A-row masking for the (rare) partial-O case.
    float am;
    const float* Wrow;
    if (FULL_O) {
        am = 1.0f;
        Wrow = Wm + (size_t)ar * C;
    } else {
        am = (ar < O) ? 1.0f : 0.0f;
        const int arc = (ar < O) ? ar : (O - 1);
        Wrow = Wm + (size_t)arc * C;
    }

    v8f acc[TN];
    #pragma unroll
    for (int t = 0; t < TN; ++t) acc[t] = (v8f){};

    const int Cmain = C & ~3;
    int c = 0;
    #pragma unroll 2
    for (; c < Cmain; c += 4) {
        const int k0 = c + 2 * half;
        // A operand: lane row ar, K-pair (k0, k0+1)   (unconditional loads)
        v2f Av;
        Av.x = am * Wrow[k0];
        Av.y = am * Wrow[k0 + 1];
        // B operand(s): coalesced 16-float runs per half-wave, TN tiles
        const float* xk0 = Xb + (size_t)k0 * P + pc;
        const float* xk1 = xk0 + P;
        float b0[TN], b1[TN];
        #pragma unroll
        for (int t = 0; t < TN; ++t) { b0[t] = xk0[16 * t]; b1[t] = xk1[16 * t]; }
        #pragma unroll
        for (int t = 0; t < TN; ++t) {
            v2f Bv = {b0[t], b1[t]};
            acc[t] = __builtin_amdgcn_wmma_f32_16x16x4_f32(false, Av, false, Bv,
                                                           (short)0, acc[t], false, false);
        }
    }
    if (c < C) {                         // tail: C % 4 channels, guarded once
        const int k0 = c + 2 * half;
        v2f Av;
        Av.x = (k0     < C) ? am * Wrow[k0]     : 0.0f;
        Av.y = (k0 + 1 < C) ? am * Wrow[k0 + 1] : 0.0f;
        #pragma unroll
        for (int t = 0; t < TN; ++t) {
            float b0 = (k0     < C) ? Xb[(size_t)k0 * P + pc + 16 * t]       : 0.0f;
            float b1 = (k0 + 1 < C) ? Xb[(size_t)(k0 + 1) * P + pc + 16 * t] : 0.0f;
            v2f Bv = {b0, b1};
            acc[t] = __builtin_amdgcn_wmma_f32_16x16x4_f32(false, Av, false, Bv,
                                                           (short)0, acc[t], false, false);
        }
    }

    // D layout: lanes 0-15 -> N=l, M=v ; lanes 16-31 -> N=l, M=v+8
    #pragma unroll
    for (int v = 0; v < 8; ++v) {
        const int o = o0 + v + 8 * half;
        if (FULL_O || o < O) {
            const float s  = gamma[o] * rsqrtf(1.0f + EPS_BN);
            const float bb = bias[o];
            const float bt = beta[o];
            float* yrow = Yb + (size_t)o * P + pc;
            #pragma unroll
            for (int t = 0; t < TN; ++t)
                yrow[16 * t] = fmaxf((acc[t][v] + bb) * s + bt, 0.0f);
        }
    }
}

// ======================================================================
// Farthest point sampling. One block per batch. xyz strided accessor:
//   coord(b,c,n) = xyz[b*bs + c*cs + n*ps]
// Emits idx[b][i] (carry-before-update, idx[0]=0) and newxyz (B,S,3).
// ======================================================================
__global__ void k_fps(const float* __restrict__ xyz, long bs, long cs, long ps,
                      int N, int npoint, int* __restrict__ idx,
                      float* __restrict__ newxyz)
{
    const int b = blockIdx.x;
    const float* X = xyz + (size_t)b * bs;
    __shared__ float dist[4096];
    __shared__ float rv[256];
    __shared__ int   ri[256];
    __shared__ int   s_far;
    const int t = threadIdx.x;

    for (int n = t; n < N; n += blockDim.x) dist[n] = 1e10f;
    if (t == 0) s_far = 0;
    __syncthreads();

    for (int it = 0; it < npoint; ++it) {
        const int far = s_far;
        const float cx = X[0*cs + (size_t)far*ps];
        const float cy = X[1*cs + (size_t)far*ps];
        const float cz = X[2*cs + (size_t)far*ps];
        if (t == 0) {
            idx[b*npoint + it] = far;
            newxyz[((size_t)b*npoint + it)*3 + 0] = cx;
            newxyz[((size_t)b*npoint + it)*3 + 1] = cy;
            newxyz[((size_t)b*npoint + it)*3 + 2] = cz;
        }
        float bv = -1.0f; int bi = 0x7fffffff;
        for (int n = t; n < N; n += blockDim.x) {
            const float dx = X[0*cs + (size_t)n*ps] - cx;
            const float dy = X[1*cs + (size_t)n*ps] - cy;
            const float dz = X[2*cs + (size_t)n*ps] - cz;
            const float d  = dx*dx + dy*dy + dz*dz;
            const float m  = fminf(dist[n], d);
            dist[n] = m;
            if (m > bv) { bv = m; bi = n; }   // strict > => smallest index wins ties
        }
        rv[t] = bv; ri[t] = bi;
        __syncthreads();
        for (int s = blockDim.x >> 1; s > 0; s >>= 1) {
            if (t < s) {
                if (rv[t+s] > rv[t] || (rv[t+s] == rv[t] && ri[t+s] < ri[t])) {
                    rv[t] = rv[t+s]; ri[t] = ri[t+s];
                }
            }
            __syncthreads();
        }
        if (t == 0) s_far = ri[0];
        __syncthreads();
    }
}

// ======================================================================
// Ball query: first `nsample` indices (ascending) with d^2 <= r^2.
// ======================================================================
__global__ void k_ball(const float* __restrict__ xyz, long bs, long cs, long ps, int N,
                       const float* __restrict__ newxyz, int S, float r2, int nsample,
                       int* __restrict__ out, int total)
{
    const int g = blockIdx.x * blockDim.x + threadIdx.x;
    if (g >= total) return;
    const int b = g / S, s = g % S;
    const float* X = xyz + (size_t)b * bs;
    const float cx = newxyz[((size_t)b*S + s)*3 + 0];
    const float cy = newxyz[((size_t)b*S + s)*3 + 1];
    const float cz = newxyz[((size_t)b*S + s)*3 + 2];
    int* o = out + ((size_t)b*S + s) * nsample;
    int cnt = 0, first = N - 1;
    for (int n = 0; n < N && cnt < nsample; ++n) {
        const float dx = X[0*cs + (size_t)n*ps] - cx;
        const float dy = X[1*cs + (size_t)n*ps] - cy;
        const float dz = X[2*cs + (size_t)n*ps] - cz;
        if (dx*dx + dy*dy + dz*dz <= r2) {
            if (cnt == 0) first = n;
            o[cnt++] = n;
        }
    }
    for (int k = cnt; k < nsample; ++k) o[k] = first;
}

// ======================================================================
// Grouping: out[b][c][k*S+s] ; c<3: centered xyz gather, else point gather.
// ======================================================================
__global__ void k_group(const float* __restrict__ xyz, long xbs, long xcs, long xps,
                        const float* __restrict__ pts, long pbs, long pcs, int D,
                        const float* __restrict__ newxyz, const int* __restrict__ idx,
                        int S, int K, float* __restrict__ out, int total)
{
    const int g = blockIdx.x * blockDim.x + threadIdx.x;
    if (g >= total) return;
    const int C = 3 + D;
    const int s = g % S;
    const int k = (g / S) % K;
    const int c = (g / (S * K)) % C;
    const int b = g / (S * K * C);
    const int n = idx[((size_t)b*S + s)*K + k];
    float v;
    if (c < 3) v = xyz[(size_t)b*xbs + (size_t)c*xcs + (size_t)n*xps]
                   - newxyz[((size_t)b*S + s)*3 + c];
    else       v = pts[(size_t)b*pbs + (size_t)(c-3)*pcs + n];
    out[(((size_t)b*C + c)*K + k)*S + s] = v;
}

// group_all (sa3): out[b][c][k] = c<3 ? xyz2[b][k][c] : pts[b][c-3][k]
__global__ void k_groupall(const float* __restrict__ xyz2, const float* __restrict__ pts,
                           int D, int K, float* __restrict__ out, int total)
{
    const int g = blockIdx.x * blockDim.x + threadIdx.x;
    if (g >= total) return;
    const int C = 3 + D;
    const int k = g % K;
    const int c = (g / K) % C;
    const int b = g / (K * C);
    out[((size_t)b*C + c)*K + k] = (c < 3) ? xyz2[((size_t)b*K + k)*3 + c]
                                           : pts[((size_t)b*D + (c-3))*K + k];
}

// ======================================================================
// Max-pool over K: in (B,O,K,S) -> out (B,O,S)
// ======================================================================
__global__ void k_maxpool(const float* __restrict__ in, float* __restrict__ out,
                          int O, int K, int S, int total)
{
    const int g = blockIdx.x * blockDim.x + threadIdx.x;
    if (g >= total) return;
    const int s = g % S;
    const int o = (g / S) % O;
    const int b = g / (S * O);
    const float* p = in + (((size_t)b*O + o)*K)*S + s;
    float m = p[0];
    for (int k = 1; k < K; ++k) m = fmaxf(m, p[(size_t)k*S]);
    out[((size_t)b*O + o)*S + s] = m;
}

// ======================================================================
// 3-NN + inverse-distance weights. xyz1 strided; xyz2 point-major (B,S2,3).
// ======================================================================
__global__ void k_knn3(const float* __restrict__ x1, long bs1, long cs1, long ps1,
                       const float* __restrict__ x2, int S2, int N1,
                       int* __restrict__ idx3, float* __restrict__ w3, int total)
{
    const int g = blockIdx.x * blockDim.x + threadIdx.x;
    if (g >= total) return;
    const int b = g / N1, n = g % N1;
    const float ax = x1[(size_t)b*bs1 + 0*cs1 + (size_t)n*ps1];
    const float ay = x1[(size_t)b*bs1 + 1*cs1 + (size_t)n*ps1];
    const float az = x1[(size_t)b*bs1 + 2*cs1 + (size_t)n*ps1];
    float d0 = FLT_MAX, d1 = FLT_MAX, d2 = FLT_MAX;
    int   i0 = 0, i1 = 0, i2 = 0;
    for (int j = 0; j < S2; ++j) {
        const float dx = ax - x2[((size_t)b*S2 + j)*3 + 0];
        const float dy = ay - x2[((size_t)b*S2 + j)*3 + 1];
        const float dz = az - x2[((size_t)b*S2 + j)*3 + 2];
        const float d  = dx*dx + dy*dy + dz*dz;
        if (d < d0)      { d2=d1; i2=i1; d1=d0; i1=i0; d0=d; i0=j; }
        else if (d < d1) { d2=d1; i2=i1; d1=d; i1=j; }
        else if (d < d2) { d2=d; i2=j; }
    }
    float w0 = 1.0f / (d0 + 1e-8f);
    float w1 = 1.0f / (d1 + 1e-8f);
    float w2 = 1.0f / (d2 + 1e-8f);
    const float inv = 1.0f / (w0 + w1 + w2);
    const size_t o = ((size_t)b*N1 + n) * 3;
    idx3[o+0] = i0; idx3[o+1] = i1; idx3[o+2] = i2;
    w3[o+0] = w0*inv; w3[o+1] = w1*inv; w3[o+2] = w2*inv;
}

// ======================================================================
// Concat [skip features ; interpolated features] -> out (B, D1+D2, N1).
// idx3 == nullptr  => S2==1 broadcast path.
// ======================================================================
__global__ void k_interp_concat(const float* __restrict__ p1, long p1bs, long p1cs, int D1,
                                const float* __restrict__ p2, int D2, int S2,
                                const int* __restrict__ idx3, const float* __restrict__ w3,
                                int N1, float* __restrict__ out, int total)
{
    const int g = blockIdx.x * blockDim.x + threadIdx.x;
    if (g >= total) return;
    const int C = D1 + D2;
    const int n = g % N1;
    const int c = (g / N1) % C;
    const int b = g / (N1 * C);
    float v;
    if (c < D1) {
        v = p1[(size_t)b*p1bs + (size_t)c*p1cs + n];
    } else {
        const float* P2 = p2 + ((size_t)b*D2 + (c - D1)) * S2;
        if (idx3 == nullptr) {
            v = P2[0];
        } else {
            const size_t o = ((size_t)b*N1 + n) * 3;
            v = w3[o+0]*P2[idx3[o+0]] + w3[o+1]*P2[idx3[o+1]] + w3[o+2]*P2[idx3[o+2]];
        }
    }
    out[((size_t)b*C + c)*N1 + n] = v;
}

// ======================================================================
// Host orchestration
// ======================================================================
extern "C" void kernel_launch(void* const* d_in, const int* in_sizes, int n_in,
                              void* d_out, int out_size, void* d_ws, size_t ws_size,
                              hipStream_t stream)
{
    (void)in_sizes; (void)n_in; (void)out_size; (void)ws_size;

    const int B = 32, N0 = 4096, S1 = 128, K1 = 32, S2n = 32, K2 = 64, K3 = 32;

    const float* xyz = (const float*)d_in[0];

    struct Layer { const float *W, *b, *beta, *gamma; int ic, oc; };
    int pi = 1;
    auto nx = [&](int ic, int oc) {
        Layer L; L.W = (const float*)d_in[pi]; L.b = (const float*)d_in[pi+1];
        L.beta = (const float*)d_in[pi+2]; L.gamma = (const float*)d_in[pi+3];
        pi += 4; L.ic = ic; L.oc = oc; return L;
    };
    // JAX pytree leaf order: dict keys sorted -> fp1, fp2, fp3, sa1, sa2, sa3
    // per-layer dict sorted -> W, b, beta, gamma
    Layer fp1[3] = { nx(70, 64),  nx(64, 32),  nx(32, 6)   };
    Layer fp2[2] = { nx(192,128), nx(128,64)               };
    Layer fp3[2] = { nx(384,128), nx(128,128)              };
    Layer sa1[3] = { nx(9, 32),   nx(32, 32),  nx(32, 64)  };
    Layer sa2[3] = { nx(67, 64),  nx(64, 64),  nx(64, 128) };
    Layer sa3[3] = { nx(131,128), nx(128,128), nx(128,256) };

    // ---- workspace layout ----
    char* wsb = (char*)d_ws;
    size_t off = 0;
    auto fal = [&](size_t n) { float* p = (float*)(wsb + off); off += n * sizeof(float); return p; };
    auto ial = [&](size_t n) { int*   p = (int*)  (wsb + off); off += n * sizeof(int);   return p; };

    int*   fpsidx1  = ial((size_t)B * S1);
    float* newxyz1  = fal((size_t)B * S1 * 3);
    int*   ballidx1 = ial((size_t)B * S1 * K1);
    int*   fpsidx2  = ial((size_t)B * S2n);
    float* newxyz2  = fal((size_t)B * S2n * 3);
    int*   ballidx2 = ial((size_t)B * S2n * K2);
    float* l1p      = fal((size_t)B * 64  * S1);      // l1_points (B,64,128)
    float* l2p      = fal((size_t)B * 128 * S2n);     // l2_points (B,128,32)
    float* l3p      = fal((size_t)B * 256 * 1);       // l3_points (B,256,1)
    float* l2new    = fal((size_t)B * 128 * S2n);     // fp3 output
    float* l1new    = fal((size_t)B * 64  * S1);      // fp2 output
    int*   knnidx   = ial((size_t)B * N0 * 3);        // sized for fp1; reused by fp2
    float* knnw     = fal((size_t)B * N0 * 3);
    float* bigA     = fal((size_t)B * 70 * N0);       // grouped / concat inputs
    float* bigB     = fal((size_t)B * 64 * N0);       // ping
    float* bigC     = fal((size_t)B * 64 * N0);       // pong

    auto conv = [&](const float* X, float* Y, const Layer& L, int P) {
        const bool fo = (L.oc % 16) == 0;
        if (P % 64 == 0) {
            dim3 g((unsigned)(P / 64), (unsigned)((L.oc + 15) / 16), (unsigned)B);
            if (fo) k_conv_bn_relu<4, true ><<<g, 32, 0, stream>>>(X, Y, L.W, L.b, L.beta, L.gamma, L.ic, L.oc, P);
            else    k_conv_bn_relu<4, false><<<g, 32, 0, stream>>>(X, Y, L.W, L.b, L.beta, L.gamma, L.ic, L.oc, P);
        } else {   // P % 32 == 0 for every remaining layer in this network
            dim3 g((unsigned)(P / 32), (unsigned)((L.oc + 15) / 16), (unsigned)B);
            if (fo) k_conv_bn_relu<2, true ><<<g, 32, 0, stream>>>(X, Y, L.W, L.b, L.beta, L.gamma, L.ic, L.oc, P);
            else    k_conv_bn_relu<2, false><<<g, 32, 0, stream>>>(X, Y, L.W, L.b, L.beta, L.gamma, L.ic, L.oc, P);
        }
    };
    auto blocks = [](int total, int bs) { return (unsigned)((total + bs - 1) / bs); };

    // ---------------- SA1: N=4096 -> S=128, K=32, r=0.02, C_in=9 ----------------
    k_fps<<<B, 256, 0, stream>>>(xyz, (long)6*N0, (long)N0, 1L, N0, S1, fpsidx1, newxyz1);
    {
        int tot = B * S1;
        k_ball<<<blocks(tot,128), 128, 0, stream>>>(xyz, (long)6*N0, (long)N0, 1L, N0,
                                                    newxyz1, S1, 0.02f*0.02f, K1, ballidx1, tot);
    }
    {
        int tot = B * 9 * K1 * S1;
        k_group<<<blocks(tot,256), 256, 0, stream>>>(xyz, (long)6*N0, (long)N0, 1L,
                                                     xyz, (long)6*N0, (long)N0, 6,
                                                     newxyz1, ballidx1, S1, K1, bigA, tot);
    }
    conv(bigA, bigB, sa1[0], K1*S1);
    conv(bigB, bigC, sa1[1], K1*S1);
    conv(bigC, bigB, sa1[2], K1*S1);
    {
        int tot = B * 64 * S1;
        k_maxpool<<<blocks(tot,256), 256, 0, stream>>>(bigB, l1p, 64, K1, S1, tot);
    }

    // ---------------- SA2: N=128 -> S=32, K=64, r=0.04, C_in=67 ----------------
    k_fps<<<B, 256, 0, stream>>>(newxyz1, (long)S1*3, 1L, 3L, S1, S2n, fpsidx2, newxyz2);
    {
        int tot = B * S2n;
        k_ball<<<blocks(tot,128), 128, 0, stream>>>(newxyz1, (long)S1*3, 1L, 3L, S1,
                                                    newxyz2, S2n, 0.04f*0.04f, K2, ballidx2, tot);
    }
    {
        int tot = B * 67 * K2 * S2n;
        k_group<<<blocks(tot,256), 256, 0, stream>>>(newxyz1, (long)S1*3, 1L, 3L,
                                                     l1p, (long)64*S1, (long)S1, 64,
                                                     newxyz2, ballidx2, S2n, K2, bigA, tot);
    }
    conv(bigA, bigB, sa2[0], K2*S2n);
    conv(bigB, bigC, sa2[1], K2*S2n);
    conv(bigC, bigB, sa2[2], K2*S2n);
    {
        int tot = B * 128 * S2n;
        k_maxpool<<<blocks(tot,256), 256, 0, stream>>>(bigB, l2p, 128, K2, S2n, tot);
    }

    // ---------------- SA3 (group_all): K=32 points, C_in=131 ----------------
    {
        int tot = B * 131 * K3;
        k_groupall<<<blocks(tot,256), 256, 0, stream>>>(newxyz2, l2p, 128, K3, bigA, tot);
    }
    conv(bigA, bigB, sa3[0], K3);
    conv(bigB, bigC, sa3[1], K3);
    conv(bigC, bigB, sa3[2], K3);
    {
        int tot = B * 256;
        k_maxpool<<<blocks(tot,256), 256, 0, stream>>>(bigB, l3p, 256, K3, 1, tot);
    }

    // ---------------- FP3: S2==1 broadcast, concat 128+256=384 -> 128 -> 128 ----------------
    {
        int tot = B * 384 * S2n;
        k_interp_concat<<<blocks(tot,256), 256, 0, stream>>>(l2p, (long)128*S2n, (long)S2n, 128,
                                                             l3p, 256, 1,
                                                             nullptr, nullptr, S2n, bigA, tot);
    }
    conv(bigA, bigB, fp3[0], S2n);
    conv(bigB, l2new, fp3[1], S2n);

    // ---------------- FP2: knn(128 pts -> 32 ref), concat 64+128=192 -> 128 -> 64 ----------------
    {
        int tot = B * S1;
        k_knn3<<<blocks(tot,128), 128, 0, stream>>>(newxyz1, (long)S1*3, 1L, 3L,
                                                    newxyz2, S2n, S1, knnidx, knnw, tot);
    }
    {
        int tot = B * 192 * S1;
        k_interp_concat<<<blocks(tot,256), 256, 0, stream>>>(l1p, (long)64*S1, (long)S1, 64,
                                                             l2new, 128, S2n,
                                                             knnidx, knnw, S1, bigA, tot);
    }
    conv(bigA, bigB, fp2[0], S1);
    conv(bigB, l1new, fp2[1], S1);

    // ---------------- FP1: knn(4096 pts -> 128 ref), concat 6+64=70 -> 64 -> 32 -> 6 ----------------
    {
        int tot = B * N0;
        k_knn3<<<blocks(tot,128), 128, 0, stream>>>(xyz, (long)6*N0, (long)N0, 1L,
                                                    newxyz1, S1, N0, knnidx, knnw, tot);
    }
    {
        int tot = B * 70 * N0;
        k_interp_concat<<<blocks(tot,256), 256, 0, stream>>>(xyz, (long)6*N0, (long)N0, 6,
                                                             l1new, 64, S1,
                                                             knnidx, knnw, N0, bigA, tot);
    }
    conv(bigA, bigB, fp1[0], N0);
    conv(bigB, bigC, fp1[1], N0);
    conv(bigC, (float*)d_out, fp1[2], N0);   // final (B,6,4096) with ReLU, per reference
}